// SetDecoder_66795331387538
// MI455X (gfx1250) — compile-verified
//
#include <hip/hip_runtime.h>
#include <math.h>

// ---------------------------------------------------------------------------
// Slot-attention + MESH Sinkhorn forward for MI455X (gfx1250, wave32).
// Heavy GEMMs (4096-row) use V_WMMA_F32_16X16X4_F32 (full f32 precision),
// software-pipelined (register double buffering) with 32x64 C tiles per wave.
// ---------------------------------------------------------------------------

namespace {

constexpr int Bc = 8, Nc = 512, Hc = 768, Sc = 15, NCc = 97;
constexpr int BN  = Bc * Nc;        // 4096
constexpr int BS  = Bc * Sc;        // 120
constexpr int BNH = BN * Hc;        // 3,145,728
constexpr int BSH = BS * Hc;        // 92,160
constexpr int BNS = BN * Sc;        // 61,440
constexpr float INV_NS = 1.0f / float(Nc * Sc);

typedef __attribute__((ext_vector_type(2))) float v2f;
typedef __attribute__((ext_vector_type(8))) float v8f;

__device__ __forceinline__ float wave_sum(float v) {
  #pragma unroll
  for (int o = 16; o > 0; o >>= 1) v += __shfl_xor(v, o, 32);
  return v;
}
__device__ __forceinline__ float block_sum256(float v, float* sred) {
  int t = threadIdx.x;
  sred[t] = v; __syncthreads();
  for (int st = 128; st > 0; st >>= 1) { if (t < st) sred[t] += sred[t + st]; __syncthreads(); }
  float r = sred[0]; __syncthreads();
  return r;
}
__device__ __forceinline__ float block_max256(float v, float* sred) {
  int t = threadIdx.x;
  sred[t] = v; __syncthreads();
  for (int st = 128; st > 0; st >>= 1) { if (t < st) sred[t] = fmaxf(sred[t], sred[t + st]); __syncthreads(); }
  float r = sred[0]; __syncthreads();
  return r;
}
__device__ __forceinline__ float gelu_exact(float x) {
  return 0.5f * x * (1.0f + erff(x * 0.70710678118654752f));
}
__device__ __forceinline__ float sigm(float x) { return 1.0f / (1.0f + expf(-x)); }

// ---- layernorm over rows of length H (blockDim=256, 3 elems/thread) -------
__global__ void k_layernorm(const float* __restrict__ in, const float* __restrict__ g,
                            const float* __restrict__ bsh, float* __restrict__ out, int rows) {
  __shared__ float sred[256];
  int row = blockIdx.x; if (row >= rows) return;
  const float* ip = in + (size_t)row * Hc;
  int t = threadIdx.x;
  float l0 = ip[t], l1 = ip[t + 256], l2 = ip[t + 512];
  float mean = block_sum256(l0 + l1 + l2, sred) / float(Hc);
  float d0 = l0 - mean, d1 = l1 - mean, d2 = l2 - mean;
  float var = block_sum256(d0 * d0 + d1 * d1 + d2 * d2, sred) / float(Hc);
  float rstd = rsqrtf(var + 1e-5f);
  float* op = out + (size_t)row * Hc;
  op[t]       = d0 * rstd * g[t]       + bsh[t];
  op[t + 256] = d1 * rstd * g[t + 256] + bsh[t + 256];
  op[t + 512] = d2 * rstd * g[t + 512] + bsh[t + 512];
}

// ---- f32 WMMA GEMM: OUT[m,j] = act(sum_k A[m,k]*W[j,k] + bias[j]) ---------
// grid = (Nout/64, M/256), block = 256 (8 waves); each wave -> 32x64 C tile
// (2 M-tiles x 4 N-tiles = 8 WMMAs per k-step from 6 b64 loads).
// Register double-buffering: prefetch chunk k+4 before WMMAs of chunk k so
// the scheduler can overlap global loads with the matrix pipe.
// A frag (16x4): lane&15 = M, (lane>>4)*2 = K-half. B frag mirrored (4x16).
// C: VGPR r -> M=r (lanes 0-15) / M=r+8 (lanes 16-31), N = lane&15.
__global__ void k_wmma_gemm(const float* __restrict__ A, const float* __restrict__ W,
                            const float* __restrict__ bias, float* __restrict__ out,
                            int K, int Nout, int act) {
  const int lane = threadIdx.x & 31;
  const int wv   = threadIdx.x >> 5;
  const int m0   = blockIdx.y * 256 + wv * 32;
  const int n0   = blockIdx.x * 64;
  const int kh   = (lane >> 4) << 1;     // 0 or 2 (lane-half K offset)
  const int l15  = lane & 15;
  // all addresses are even-float offsets -> 8B aligned -> b64 loads
  const float* a0p = A + (size_t)(m0      + l15) * K + kh;
  const float* a1p = A + (size_t)(m0 + 16 + l15) * K + kh;
  const float* w0p = W + (size_t)(n0      + l15) * K + kh;
  const float* w1p = W + (size_t)(n0 + 16 + l15) * K + kh;
  const float* w2p = W + (size_t)(n0 + 32 + l15) * K + kh;
  const float* w3p = W + (size_t)(n0 + 48 + l15) * K + kh;

  v8f acc[8] = {};   // [mtile*4 + ntile]

  // prefetch chunk 0
  v2f a0n = *(const v2f*)(a0p);
  v2f a1n = *(const v2f*)(a1p);
  v2f b0n = *(const v2f*)(w0p);
  v2f b1n = *(const v2f*)(w1p);
  v2f b2n = *(const v2f*)(w2p);
  v2f b3n = *(const v2f*)(w3p);

  for (int k = 0; k < K; k += 4) {
    v2f a0 = a0n, a1 = a1n, b0 = b0n, b1 = b1n, b2 = b2n, b3 = b3n;
    int kn = k + 4;
    if (kn < K) {                       // uniform branch: EXEC stays all-ones
      a0n = *(const v2f*)(a0p + kn);
      a1n = *(const v2f*)(a1p + kn);
      b0n = *(const v2f*)(w0p + kn);
      b1n = *(const v2f*)(w1p + kn);
      b2n = *(const v2f*)(w2p + kn);
      b3n = *(const v2f*)(w3p + kn);
    }
    acc[0] = __builtin_amdgcn_wmma_f32_16x16x4_f32(false, a0, false, b0, (short)0, acc[0], false, false);
    acc[1] = __builtin_amdgcn_wmma_f32_16x16x4_f32(false, a0, false, b1, (short)0, acc[1], false, false);
    acc[2] = __builtin_amdgcn_wmma_f32_16x16x4_f32(false, a0, false, b2, (short)0, acc[2], false, false);
    acc[3] = __builtin_amdgcn_wmma_f32_16x16x4_f32(false, a0, false, b3, (short)0, acc[3], false, false);
    acc[4] = __builtin_amdgcn_wmma_f32_16x16x4_f32(false, a1, false, b0, (short)0, acc[4], false, false);
    acc[5] = __builtin_amdgcn_wmma_f32_16x16x4_f32(false, a1, false, b1, (short)0, acc[5], false, false);
    acc[6] = __builtin_amdgcn_wmma_f32_16x16x4_f32(false, a1, false, b2, (short)0, acc[6], false, false);
    acc[7] = __builtin_amdgcn_wmma_f32_16x16x4_f32(false, a1, false, b3, (short)0, acc[7], false, false);
  }

  const int rofs = (lane >> 4) * 8;
  #pragma unroll
  for (int mt = 0; mt < 2; ++mt) {
    #pragma unroll
    for (int r = 0; r < 8; ++r) {
      const int mm = m0 + 16 * mt + rofs + r;
      float* orow = out + (size_t)mm * Nout + n0 + l15;
      #pragma unroll
      for (int nt = 0; nt < 4; ++nt) {
        float tvl = acc[mt * 4 + nt][r] + (bias ? bias[n0 + 16 * nt + l15] : 0.f);
        if (act) tvl = gelu_exact(tvl);
        orow[16 * nt] = tvl;
      }
    }
  }
}

// ---- small LDS-tiled GEMM for slot-sized (M=120) ops ----------------------
__global__ void k_sgemm(const float* __restrict__ A, const float* __restrict__ W,
                        const float* __restrict__ bias, float* __restrict__ out,
                        int M, int Nout, int K, int act, int accflag) {
  __shared__ float As[16][17];
  __shared__ float Ws[16][17];
  int tx = threadIdx.x, ty = threadIdx.y;
  int m = blockIdx.y * 16 + ty;
  int j = blockIdx.x * 16 + tx;
  float acc = 0.f;
  for (int kk = 0; kk < K; kk += 16) {
    As[ty][tx] = (m < M) ? A[(size_t)m * K + kk + tx] : 0.f;
    int jr = blockIdx.x * 16 + ty;
    Ws[ty][tx] = (jr < Nout) ? W[(size_t)jr * K + kk + tx] : 0.f;
    __syncthreads();
    #pragma unroll
    for (int i = 0; i < 16; ++i) acc += As[ty][i] * Ws[tx][i];
    __syncthreads();
  }
  if (m < M && j < Nout) {
    float vv = acc + (bias ? bias[j] : 0.f);
    if (act) vv = gelu_exact(vv);
    if (accflag) vv += out[(size_t)m * Nout + j];
    out[(size_t)m * Nout + j] = vv;
  }
}

// ---- row L2-norm (wave per row) -------------------------------------------
__global__ void k_rownorm(const float* __restrict__ A, float* __restrict__ out, int rows, int cols) {
  int wv = (blockIdx.x * blockDim.x + threadIdx.x) >> 5;
  int lane = threadIdx.x & 31;
  if (wv >= rows) return;
  const float* p = A + (size_t)wv * cols;
  float s = 0.f;
  for (int h = lane; h < cols; h += 32) { float t = p[h]; s += t * t; }
  s = wave_sum(s);
  if (lane == 0) out[wv] = fmaxf(sqrtf(s), 1e-8f);
}

// ---- row dot with a shared vector (wave per row) --------------------------
__global__ void k_rowdot(const float* __restrict__ A, const float* __restrict__ w,
                         float* __restrict__ out, int rows, int cols) {
  int wv = (blockIdx.x * blockDim.x + threadIdx.x) >> 5;
  int lane = threadIdx.x & 31;
  if (wv >= rows) return;
  const float* p = A + (size_t)wv * cols;
  float s = 0.f;
  for (int h = lane; h < cols; h += 32) s += p[h] * w[h];
  s = wave_sum(s);
  if (lane == 0) out[wv] = s;
}

// ---- log_a = log(clip(S*softmax_n(score), 1e-30)) (block per b, n=512) ----
__global__ void k_logits_n(const float* __restrict__ sc, float* __restrict__ log_a) {
  __shared__ float sred[256];
  int b = blockIdx.x, t = threadIdx.x;
  const float* p = sc + (size_t)b * Nc;
  float x0 = p[t], x1 = p[t + 256];
  float mx = block_max256(fmaxf(x0, x1), sred);
  float e0 = expf(x0 - mx), e1 = expf(x1 - mx);
  float sm = block_sum256(e0 + e1, sred);
  float* q = log_a + (size_t)b * Nc;
  q[t]       = logf(fmaxf(float(Sc) * e0 / sm, 1e-30f));
  q[t + 256] = logf(fmaxf(float(Sc) * e1 / sm, 1e-30f));
}

// ---- log_b over s (thread per b, s=15) ------------------------------------
__global__ void k_logits_s(const float* __restrict__ sc, float* __restrict__ log_b) {
  int b = blockIdx.x * blockDim.x + threadIdx.x;
  if (b >= Bc) return;
  const float* p = sc + (size_t)b * Sc;
  float mx = -1e30f;
  for (int s = 0; s < Sc; ++s) mx = fmaxf(mx, p[s]);
  float sm = 0.f;
  for (int s = 0; s < Sc; ++s) sm += expf(p[s] - mx);
  for (int s = 0; s < Sc; ++s)
    log_b[b * Sc + s] = logf(fmaxf(float(Sc) * expf(p[s] - mx) / sm, 1e-30f));
}

__global__ void k_init_slots(const float* __restrict__ mu, const float* __restrict__ lsig,
                             const float* __restrict__ ini, float* __restrict__ slots) {
  int i = blockIdx.x * blockDim.x + threadIdx.x;
  if (i >= BSH) return;
  int h = i % Hc;
  slots[i] = mu[h] + expf(lsig[h]) * ini[i];
}

__global__ void k_copy(const float* __restrict__ a, float* __restrict__ b, int n) {
  int i = blockIdx.x * blockDim.x + threadIdx.x;
  if (i < n) b[i] = a[i];
}
__global__ void k_fill(float* __restrict__ a, float v, int n) {
  int i = blockIdx.x * blockDim.x + threadIdx.x;
  if (i < n) a[i] = v;
}

// ---- cost C[b,n,s] = S*(1 - k.q/(|k||q|)) + 0.001*noise (wave per (b,n)) --
__global__ void k_cost(const float* __restrict__ kk, const float* __restrict__ q,
                       const float* __restrict__ knrm, const float* __restrict__ qnrm,
                       const float* __restrict__ noise, float* __restrict__ C) {
  int wv = (blockIdx.x * blockDim.x + threadIdx.x) >> 5;   // b*N + n
  int lane = threadIdx.x & 31;
  if (wv >= BN) return;
  int b = wv / Nc;
  const float* kp = kk + (size_t)wv * Hc;
  float kreg[24];
  #pragma unroll
  for (int i = 0; i < 24; ++i) kreg[i] = kp[lane + 32 * i];
  float kn = knrm[wv];
  for (int s = 0; s < Sc; ++s) {
    const float* qp = q + (size_t)(b * Sc + s) * Hc;
    float d = 0.f;
    #pragma unroll
    for (int i = 0; i < 24; ++i) d += kreg[i] * qp[lane + 32 * i];
    d = wave_sum(d);
    if (lane == 0) {
      float c = float(Sc) * (1.0f - d / (kn * qnrm[b * Sc + s]));
      C[(size_t)wv * Sc + s] = c + 0.001f * noise[(size_t)wv * Sc + s];
    }
  }
}

// ---- sinkhorn u-update (thread per (b,n)) ---------------------------------
__global__ void k_u_update(const float* __restrict__ C, const float* __restrict__ log_a,
                           const float* __restrict__ vin, float* __restrict__ uout) {
  int t = blockIdx.x * blockDim.x + threadIdx.x;
  if (t >= BN) return;
  int b = t / Nc;
  float vals[Sc];
  float mx = -1e30f;
  for (int s = 0; s < Sc; ++s) {
    float v = -C[(size_t)t * Sc + s] + vin[b * Sc + s];
    vals[s] = v; mx = fmaxf(mx, v);
  }
  float sm = 0.f;
  for (int s = 0; s < Sc; ++s) sm += expf(vals[s] - mx);
  uout[t] = log_a[t] - (mx + logf(sm));
}

// ---- sinkhorn v-update (block per (b,s), reduce over n=512) ---------------
__global__ void k_v_update(const float* __restrict__ C, const float* __restrict__ log_b,
                           const float* __restrict__ uin, float* __restrict__ vout) {
  __shared__ float sred[256];
  int bs = blockIdx.x; int b = bs / Sc, s = bs % Sc;
  int n0 = threadIdx.x, n1 = threadIdx.x + 256;
  float x0 = -C[(size_t)(b * Nc + n0) * Sc + s] + uin[b * Nc + n0];
  float x1 = -C[(size_t)(b * Nc + n1) * Sc + s] + uin[b * Nc + n1];
  float mx = block_max256(fmaxf(x0, x1), sred);
  float sm = block_sum256(expf(x0 - mx) + expf(x1 - mx), sred);
  if (threadIdx.x == 0) vout[bs] = log_b[bs] - (mx + logf(sm));
}

// ---- entropy-gradient seed: G wrt logT, gp=G, gu=sum_s G ------------------
__global__ void k_gradinit(const float* __restrict__ C, const float* __restrict__ u5,
                           const float* __restrict__ v5, float* __restrict__ gp,
                           float* __restrict__ gu) {
  int t = blockIdx.x * blockDim.x + threadIdx.x;
  if (t >= BN) return;
  int b = t / Nc;
  float uu = u5[t], acc = 0.f;
  for (int s = 0; s < Sc; ++s) {
    float lt = -C[(size_t)t * Sc + s] + uu + v5[b * Sc + s];
    float T = expf(lt);
    float G = (T >= 1e-20f && T <= 1.0f) ? (-(1.0f + lt) * T * INV_NS) : 0.0f;
    gp[(size_t)t * Sc + s] = G; acc += G;
  }
  gu[t] = acc;
}

// ---- gv = sum_n gp (block per (b,s)) --------------------------------------
__global__ void k_gv_init(const float* __restrict__ gp, float* __restrict__ gv) {
  __shared__ float sred[256];
  int bs = blockIdx.x; int b = bs / Sc, s = bs % Sc;
  float a = gp[(size_t)(b * Nc + threadIdx.x) * Sc + s] +
            gp[(size_t)(b * Nc + threadIdx.x + 256) * Sc + s];
  float sm = block_sum256(a, sred);
  if (threadIdx.x == 0) gv[bs] = sm;
}

// ---- backward of v_t = log_b - LSE_n(p+u_t);  w = exp(p+u_t-log_b+v_t) ----
__global__ void k_bwd_v(const float* __restrict__ C, const float* __restrict__ ut,
                        const float* __restrict__ vt, const float* __restrict__ log_b,
                        const float* __restrict__ gv, float* __restrict__ gp,
                        float* __restrict__ gu, int keep) {
  int t = blockIdx.x * blockDim.x + threadIdx.x;
  if (t >= BN) return;
  int b = t / Nc;
  float uu = ut[t], acc = 0.f;
  for (int s = 0; s < Sc; ++s) {
    float w = expf(-C[(size_t)t * Sc + s] + uu - log_b[b * Sc + s] + vt[b * Sc + s]);
    float g = gv[b * Sc + s] * w;
    gp[(size_t)t * Sc + s] -= g;
    acc -= g;
  }
  gu[t] = (keep ? gu[t] : 0.f) + acc;
}

// ---- backward of u_t = log_a - LSE_s(p+v_{t-1}); m = exp(p+v_{t-1}-log_a+u_t)
__global__ void k_bwd_u(const float* __restrict__ C, const float* __restrict__ vprev,
                        const float* __restrict__ ut, const float* __restrict__ log_a,
                        const float* __restrict__ gu, float* __restrict__ gp,
                        float* __restrict__ gv) {
  __shared__ float sred[256];
  int bs = blockIdx.x; int b = bs / Sc, s = bs % Sc;
  float vp = vprev[bs];
  float part = 0.f;
  for (int nn = threadIdx.x; nn < Nc; nn += 256) {
    int t = b * Nc + nn;
    float m = expf(-C[(size_t)t * Sc + s] + vp - log_a[t] + ut[t]);
    float g = gu[t] * m;
    gp[(size_t)t * Sc + s] -= g;
    part -= g;
  }
  float sm = block_sum256(part, sred);
  if (threadIdx.x == 0) gv[bs] = sm;
}

// ---- per-batch grad norm of (gp + 1e-20) ----------------------------------
__global__ void k_gnorm(const float* __restrict__ gp, float* __restrict__ bn) {
  __shared__ float sred[256];
  int b = blockIdx.x;
  float acc = 0.f;
  for (int i = threadIdx.x; i < Nc * Sc; i += 256) {
    float t = gp[(size_t)b * Nc * Sc + i] + 1e-20f;
    acc += t * t;
  }
  float sm = block_sum256(acc, sred);
  if (threadIdx.x == 0) bn[b] = sqrtf(sm);
}

__global__ void k_c_update(float* __restrict__ C, const float* __restrict__ gp,
                           const float* __restrict__ bn) {
  int i = blockIdx.x * blockDim.x + threadIdx.x;
  if (i >= BNS) return;
  int b = i / (Nc * Sc);
  C[i] -= (gp[i] + 1e-20f) / fmaxf(bn[b], 1e-12f);   // MESH_LR = 1.0
}

__global__ void k_attn(const float* __restrict__ C, const float* __restrict__ u,
                       const float* __restrict__ v, float* __restrict__ attn) {
  int i = blockIdx.x * blockDim.x + threadIdx.x;
  if (i >= BNS) return;
  int t = i / Sc, s = i % Sc, b = t / Nc;
  attn[i] = expf(-C[i] + u[t] + v[b * Sc + s]);
}

// ---- updates[b,s,h] = sum_n attn[b,n,s] * v[b,n,h] (block per (b,s)) ------
__global__ void k_updates(const float* __restrict__ attn, const float* __restrict__ vmat,
                          float* __restrict__ upd) {
  int bs = blockIdx.x; int b = bs / Sc, s = bs % Sc;
  int h0 = threadIdx.x, h1 = threadIdx.x + 256, h2 = threadIdx.x + 512;
  float a0 = 0.f, a1 = 0.f, a2 = 0.f;
  for (int nn = 0; nn < Nc; ++nn) {
    float a = attn[(size_t)(b * Nc + nn) * Sc + s];
    const float* vp = vmat + (size_t)(b * Nc + nn) * Hc;
    a0 += a * vp[h0]; a1 += a * vp[h1]; a2 += a * vp[h2];
  }
  float* op = upd + (size_t)bs * Hc;
  op[h0] = a0; op[h1] = a1; op[h2] = a2;
}

// ---- GRU gate combine -----------------------------------------------------
__global__ void k_gru(const float* __restrict__ gi, const float* __restrict__ gh,
                      const float* __restrict__ hprev, float* __restrict__ out) {
  int i = blockIdx.x * blockDim.x + threadIdx.x;
  if (i >= BSH) return;
  int row = i / Hc, h = i % Hc;
  const float* gir = gi + (size_t)row * 3 * Hc;
  const float* ghr = gh + (size_t)row * 3 * Hc;
  float r = sigm(gir[h] + ghr[h]);
  float z = sigm(gir[Hc + h] + ghr[Hc + h]);
  float nn = tanhf(gir[2 * Hc + h] + r * ghr[2 * Hc + h]);
  out[i] = (1.f - z) * nn + z * hprev[i];
}

// ---- fused metric: out[b,s,n] = sum_h tanh(sl[b,s,h]+en[b,n,h])*w3[h] -----
__global__ void k_metric(const float* __restrict__ sl, const float* __restrict__ en,
                         const float* __restrict__ w3, float* __restrict__ out) {
  int wv = (blockIdx.x * blockDim.x + threadIdx.x) >> 5;
  int lane = threadIdx.x & 31;
  if (wv >= Bc * Sc * Nc) return;
  int b = wv / (Sc * Nc), rem = wv % (Sc * Nc);
  int s = rem / Nc, n = rem % Nc;
  const float* sp = sl + (size_t)(b * Sc + s) * Hc;
  const float* ep = en + (size_t)(b * Nc + n) * Hc;
  float acc = 0.f;
  for (int h = lane; h < Hc; h += 32) acc += tanhf(sp[h] + ep[h]) * w3[h];
  acc = wave_sum(acc);
  if (lane == 0) out[wv] = acc;
}

} // namespace

// ---------------------------------------------------------------------------
// Input order: 0 enc, 1 mask(unused), 2 noise, 3 slots_init, then params:
// 4 ln_in_g 5 ln_in_b 6 ln_slots_g 7 ln_slots_b 8 ln_mlp_g 9 ln_mlp_b
// 10 Wq 11 Wk 12 Wv 13 gru_wih 14 gru_whh 15 gru_bih 16 gru_bhh
// 17 mlp_W1 18 mlp_b1 19 mlp_W2 20 mlp_b2 21 sm_W1 22 sm_b1 23 sm_W2
// 24 im_W1 25 im_b1 26 im_W2 27 slots_mu 28 slots_log_sigma 29 cls_W 30 cls_b
// 31.. metric groups of 5 (W1,b1,W2,b2,W3) for hs,he,ts,te.
// ---------------------------------------------------------------------------
extern "C" void kernel_launch(void* const* d_in, const int* in_sizes, int n_in,
                              void* d_out, int out_size, void* d_ws, size_t ws_size,
                              hipStream_t stream) {
  (void)in_sizes; (void)n_in; (void)out_size; (void)ws_size;
  auto P = [&](int i) { return (const float*)d_in[i]; };
  const float* enc   = P(0);
  const float* noise = P(2);
  const float* sinit = P(3);

  float* ws = (float*)d_ws;
  size_t off = 0;
  auto alloc = [&](size_t n) { float* p = ws + off; off += n; return p; };
  float* x     = alloc(BNH);
  float* kbuf  = alloc(BNH);
  float* vbuf  = alloc(BNH);
  float* big   = alloc((size_t)2 * BNH);   // im hidden; later en_lin
  float* log_a = alloc(BN);
  float* score = alloc(BN);
  float* knrm  = alloc(BN);
  float* slots = alloc(BSH);
  float* sprev = alloc(BSH);
  float* sn    = alloc(BSH);
  float* qbuf  = alloc(BSH);
  float* upd   = alloc(BSH);
  float* sllin = alloc(BSH);
  float* hsm   = alloc((size_t)BS * 2 * Hc);
  float* gi    = alloc((size_t)BS * 3 * Hc);
  float* gh    = alloc((size_t)BS * 3 * Hc);
  float* s_sm  = alloc(BS);
  float* log_b = alloc(BS);
  float* qnrm  = alloc(BS);
  float* Cb    = alloc(BNS);
  float* gp    = alloc(BNS);               // grad / attn scratch
  float* uh    = alloc((size_t)5 * BN);    // u history (5 sinkhorn iters)
  float* vh    = alloc((size_t)5 * BS);    // v history
  float* v_p   = alloc(BS);
  float* gu    = alloc(BN);
  float* gv    = alloc(BS);
  float* bn    = alloc(Bc);

  const dim3 t16(16, 16);
  const int gY = (BS + 15) / 16;           // 8 row-blocks for M=120
  const dim3 gemmKV(Hc / 64, BN / 256);    // (12, 16)
  const dim3 gemmIM(2 * Hc / 64, BN / 256);

  // ---- prologue (iteration-invariant) ----
  k_layernorm<<<BN, 256, 0, stream>>>(enc, P(4), P(5), x, BN);
  k_wmma_gemm<<<gemmKV, 256, 0, stream>>>(x, P(11), nullptr, kbuf, Hc, Hc, 0);
  k_wmma_gemm<<<gemmKV, 256, 0, stream>>>(x, P(12), nullptr, vbuf, Hc, Hc, 0);
  k_rownorm<<<BN / 8, 256, 0, stream>>>(kbuf, knrm, BN, Hc);
  k_wmma_gemm<<<gemmIM, 256, 0, stream>>>(x, P(24), P(25), big, Hc, 2 * Hc, 1);
  k_rowdot<<<BN / 8, 256, 0, stream>>>(big, P(26), score, BN, 2 * Hc);
  k_logits_n<<<Bc, 256, 0, stream>>>(score, log_a);
  k_init_slots<<<(BSH + 255) / 256, 256, 0, stream>>>(P(27), P(28), sinit, slots);

  for (int it = 0; it < 3; ++it) {
    k_copy<<<(BSH + 255) / 256, 256, 0, stream>>>(slots, sprev, BSH);
    k_layernorm<<<BS, 256, 0, stream>>>(slots, P(6), P(7), sn, BS);
    k_sgemm<<<dim3(2 * Hc / 16, gY), t16, 0, stream>>>(sn, P(21), P(22), hsm, BS, 2 * Hc, Hc, 1, 0);
    k_rowdot<<<BS / 8, 256, 0, stream>>>(hsm, P(23), s_sm, BS, 2 * Hc);
    k_logits_s<<<1, 32, 0, stream>>>(s_sm, log_b);
    k_sgemm<<<dim3(Hc / 16, gY), t16, 0, stream>>>(sn, P(10), nullptr, qbuf, BS, Hc, Hc, 0, 0);
    k_rownorm<<<BS / 8, 256, 0, stream>>>(qbuf, qnrm, BS, Hc);
    k_cost<<<BN / 8, 256, 0, stream>>>(kbuf, qbuf, knrm, qnrm, noise, Cb);
    k_fill<<<1, 128, 0, stream>>>(v_p, 0.f, BS);

    // ---- minimize_entropy: 4 MESH iterations ----
    for (int mesh = 0; mesh < 4; ++mesh) {
      const float* vprev = v_p;
      for (int t = 0; t < 5; ++t) {            // forward sinkhorn, keep history
        k_u_update<<<BN / 256, 256, 0, stream>>>(Cb, log_a, vprev, uh + t * BN);
        k_v_update<<<BS, 256, 0, stream>>>(Cb, log_b, uh + t * BN, vh + t * BS);
        vprev = vh + t * BS;
      }
      k_gradinit<<<BN / 256, 256, 0, stream>>>(Cb, uh + 4 * BN, vh + 4 * BS, gp, gu);
      k_gv_init<<<BS, 256, 0, stream>>>(gp, gv);
      for (int t = 4; t >= 0; --t) {           // backward through sinkhorn
        k_bwd_v<<<BN / 256, 256, 0, stream>>>(Cb, uh + t * BN, vh + t * BS, log_b, gv, gp, gu,
                                              (t == 4) ? 1 : 0);
        k_bwd_u<<<BS, 256, 0, stream>>>(Cb, (t == 0) ? v_p : (vh + (t - 1) * BS),
                                        uh + t * BN, log_a, gu, gp, gv);
      }
      k_gnorm<<<Bc, 256, 0, stream>>>(gp, bn);
      k_c_update<<<(BNS + 255) / 256, 256, 0, stream>>>(Cb, gp, bn);
      k_copy<<<1, 128, 0, stream>>>(vh + 4 * BS, v_p, BS);   // carry v to next mesh iter
    }

    // ---- final sinkhorn on updated C, v0 = returned v ----
    {
      const float* vprev = v_p;
      for (int t = 0; t < 5; ++t) {
        k_u_update<<<BN / 256, 256, 0, stream>>>(Cb, log_a, vprev, uh);
        k_v_update<<<BS, 256, 0, stream>>>(Cb, log_b, uh, vh);
        vprev = vh;
      }
    }
    k_attn<<<(BNS + 255) / 256, 256, 0, stream>>>(Cb, uh, vh, gp);
    k_updates<<<BS, 256, 0, stream>>>(gp, vbuf, upd);

    // ---- GRU ----
    k_sgemm<<<dim3(3 * Hc / 16, gY), t16, 0, stream>>>(upd,   P(13), P(15), gi, BS, 3 * Hc, Hc, 0, 0);
    k_sgemm<<<dim3(3 * Hc / 16, gY), t16, 0, stream>>>(sprev, P(14), P(16), gh, BS, 3 * Hc, Hc, 0, 0);
    k_gru<<<(BSH + 255) / 256, 256, 0, stream>>>(gi, gh, sprev, slots);

    // ---- residual MLP ----
    k_layernorm<<<BS, 256, 0, stream>>>(slots, P(8), P(9), sn, BS);
    k_sgemm<<<dim3(2 * Hc / 16, gY), t16, 0, stream>>>(sn, P(17), P(18), hsm, BS, 2 * Hc, Hc, 1, 0);
    k_sgemm<<<dim3(Hc / 16, gY), t16, 0, stream>>>(hsm, P(19), P(20), slots, BS, Hc, 2 * Hc, 0, 1);
  }

  // ---- outputs ----
  float* outp = (float*)d_out;
  k_sgemm<<<dim3((NCc + 15) / 16, gY), t16, 0, stream>>>(slots, P(29), P(30), outp, BS, NCc, Hc, 0, 0);
  float* mo = outp + (size_t)BS * NCc;
  for (int mi = 0; mi < 4; ++mi) {
    int base = 31 + 5 * mi;   // W1, b1, W2, b2, W3
    k_sgemm<<<dim3(Hc / 16, gY), t16, 0, stream>>>(slots, P(base), P(base + 1), sllin, BS, Hc, Hc, 0, 0);
    k_wmma_gemm<<<gemmKV, 256, 0, stream>>>(enc, P(base + 2), P(base + 3), big, Hc, Hc, 0);
    k_metric<<<(Bc * Sc * Nc) / 8, 256, 0, stream>>>(sllin, big, P(base + 4), mo);
    mo += (size_t)Bc * Sc * Nc;
  }
}